// GPT2NonResidualAttention_43490838839704
// MI455X (gfx1250) — compile-verified
//
#include <hip/hip_runtime.h>
#include <hip/hip_bf16.h>
#include <math.h>

// Problem constants (from reference): B=4, S=1024, D=1024, H=16, HD=64
#define B_   4
#define S_   1024
#define D_   1024
#define H_   16
#define HD_  64
#define SCALE_ 0.125f          // 1/sqrt(64)
#define MASKED_BIAS_ (-10000.0f)

typedef float v2f __attribute__((ext_vector_type(2)));
typedef float v8f __attribute__((ext_vector_type(8)));
typedef unsigned int v4u __attribute__((ext_vector_type(4)));
typedef int v4i __attribute__((ext_vector_type(4)));
typedef int v8i __attribute__((ext_vector_type(8)));

#if defined(__has_builtin)
#if __has_builtin(__builtin_amdgcn_tensor_load_to_lds)
#define HAVE_TDM 1
#endif
#endif
#ifndef HAVE_TDM
#define HAVE_TDM 0
#endif

#define STR2(x) #x
#define STR(x) STR2(x)
#if HAVE_TDM
#pragma message("CDNA5-PROBE: tensor_load_to_lds builtin AVAILABLE (clang " STR(__clang_major__) ") - TDM path enabled")
#else
#pragma message("CDNA5-PROBE: tensor_load_to_lds builtin NOT available (clang " STR(__clang_major__) ") - cooperative-load fallback")
#endif

// Exact-fp32 matrix core op: D(16x16) = A(16x4) * B(4x16) + C
__device__ __forceinline__ v8f wmma_f32_16x16x4(v2f a, v2f b, v8f c) {
  return __builtin_amdgcn_wmma_f32_16x16x4_f32(
      /*neg_a=*/false, a, /*neg_b=*/false, b,
      /*c_mod=*/(short)0, c, /*reuse_a=*/false, /*reuse_b=*/false);
}

#if HAVE_TDM
// Issue a TDM 2D tile load Global -> LDS.  tile_w elems per row, tile_h rows,
// row_stride in elements, fp32 data.  Optional LDS row padding via pad codes.
__device__ __forceinline__ void tdm_load_2d(unsigned lds_addr, const void* gptr,
                                            unsigned tile_w, unsigned tile_h,
                                            unsigned long long row_stride,
                                            unsigned pad_interval_code,
                                            unsigned pad_amount_code,
                                            bool pad_en) {
  unsigned long long ga = (unsigned long long)gptr;
  v4u g0;
  g0.x = 1u;                                        // count=1, user descriptor
  g0.y = lds_addr;                                  // LDS byte address
  g0.z = (unsigned)ga;                              // global_addr[31:0]
  g0.w = (unsigned)((ga >> 32) & 0x1FFFFFFu)        // global_addr[56:32]
         | (2u << 30);                              // type = 2 ("image")
  v8i g1;
  unsigned w0 = (2u << 16);                         // data_size = 4B
  if (pad_en)
    w0 |= (1u << 20) | (pad_interval_code << 22) | (pad_amount_code << 25);
  g1[0] = (int)w0;
  g1[1] = (int)((tile_w & 0xFFFFu) << 16);                      // tensor_dim0 lo
  g1[2] = (int)(((tile_w >> 16) & 0xFFFFu) | ((tile_h & 0xFFFFu) << 16));
  g1[3] = (int)(((tile_h >> 16) & 0xFFFFu) | ((tile_w & 0xFFFFu) << 16)); // tile_dim0
  g1[4] = (int)(tile_h & 0xFFFFu);                              // tile_dim1, dim2=0
  g1[5] = (int)(unsigned)(row_stride & 0xFFFFFFFFull);          // dim0_stride lo32
  g1[6] = (int)(unsigned)((row_stride >> 32) & 0xFFFFull);      // dim0_stride hi16
  g1[7] = 0;
  v4i gz = {0, 0, 0, 0};
#if __clang_major__ >= 23
  v8i gz8 = {0, 0, 0, 0, 0, 0, 0, 0};
  __builtin_amdgcn_tensor_load_to_lds(g0, g1, gz, gz, gz8, 0);
#else
  __builtin_amdgcn_tensor_load_to_lds(g0, g1, gz, gz, 0);
#endif
}
#endif

// ---------------------------------------------------------------------------
// GEMM: C[M x N] = A[M x K] * W[K x N] + bias, fp32, WMMA f32 16x16x4.
// BM=128 (8 waves x 16 rows), BN=64, BK=32. Double-buffered LDS.
// A tile staged by the Tensor Data Mover (padded rows, stride 36 floats);
// W tile staged pair-interleaved so each B fragment is one ds_load_b64.
// ---------------------------------------------------------------------------
#define BM 128
#define BN 64
#define BK 32
#define AS_STRIDE 36        // 32 + 4 pad (TDM pad codes 4/3); 16B aligned rows
#define WP_STRIDE 160       // floats per k-pair row; 160%64==32 -> halves in
                            // disjoint LDS bank sets, conflict-free b64 reads

__global__ __launch_bounds__(256)
void gemm_bias_f32(const float* __restrict__ A, const float* __restrict__ W,
                   const float* __restrict__ bias, float* __restrict__ C,
                   int M, int N, int K) {
  __shared__ float As[2][BM * AS_STRIDE];           // 2 x 18 KB
  __shared__ float Ws[2][(BK / 2) * WP_STRIDE];     // 2 x 10 KB

  const int tid  = threadIdx.x;
  const int lane = tid & 31;
  const int wave = tid >> 5;
  const int ln   = lane & 15;
  const int half = lane >> 4;

  const int rowBase = blockIdx.y * BM;
  const int colBase = blockIdx.x * BN;
  const int wr = wave * 16;

  v8f acc[4] = {};

  // --- staging helpers -----------------------------------------------------
  auto stageA = [&](int k0, int p) {
#if HAVE_TDM
    if (wave == 0)
      tdm_load_2d((unsigned)(unsigned long long)(&As[p][0]),
                  &A[(size_t)rowBase * K + k0],
                  /*tile_w=*/BK, /*tile_h=*/BM, /*row_stride=*/(unsigned long long)K,
                  /*pad_interval 32 dw=*/4u, /*pad_amount 4 dw=*/3u, true);
#else
    #pragma unroll
    for (int t = 0; t < 4; ++t) {
      int s  = tid + t * 256;
      int r  = s >> 3;
      int c4 = (s & 7) << 2;
      float4 v = *reinterpret_cast<const float4*>(
          &A[(size_t)(rowBase + r) * K + k0 + c4]);
      float* dst = &As[p][r * AS_STRIDE + c4];
      dst[0] = v.x; dst[1] = v.y; dst[2] = v.z; dst[3] = v.w;
    }
#endif
  };
  auto stageW = [&](int k0, int p) {
    #pragma unroll
    for (int t = 0; t < 2; ++t) {
      int s  = tid + t * 256;
      int r  = s >> 4;            // k row 0..31
      int c4 = (s & 15) << 2;     // col 0..60
      float4 v = *reinterpret_cast<const float4*>(
          &W[(size_t)(k0 + r) * N + colBase + c4]);
      // pair-interleave: Ws[k/2][col][2] -> B frag = one aligned b64
      float* dst = &Ws[p][(r >> 1) * WP_STRIDE + c4 * 2 + (r & 1)];
      dst[0] = v.x; dst[2] = v.y; dst[4] = v.z; dst[6] = v.w;
    }
  };

  // --- prologue: stage first tile -----------------------------------------
  stageA(0, 0);
  stageW(0, 0);

  int p = 0;
  for (int k0 = 0; k0 < K; k0 += BK, p ^= 1) {
#if HAVE_TDM
    if (wave == 0) __builtin_amdgcn_s_wait_tensorcnt(0);
#endif
    __syncthreads();                 // tile p ready for everyone
    if (k0 + BK < K) {               // overlap: stage next tile during compute
      stageA(k0 + BK, p ^ 1);
      stageW(k0 + BK, p ^ 1);
    }
    // prefetch tile i+2 into GL2 (global_prefetch_b8) so tile i+1's DMA hits
    {
      int k2 = k0 + 2 * BK;
      if (k2 < K) {
        if (tid < BM) {
          __builtin_prefetch(&A[(size_t)(rowBase + tid) * K + k2], 0, 0);
        } else {
          int r   = (tid - BM) >> 1;       // 0..63 W rows
          int off = ((tid - BM) & 1) * 32; // two 128B lines per 256B row
          __builtin_prefetch(&W[(size_t)(k2 + r) * N + colBase + off], 0, 0);
        }
      }
    }
    #pragma unroll
    for (int kk = 0; kk < BK / 4; ++kk) {
      const int ka = 4 * kk + 2 * half;
      v2f a = *reinterpret_cast<const v2f*>(&As[p][(wr + ln) * AS_STRIDE + ka]);
      #pragma unroll
      for (int nc = 0; nc < 4; ++nc) {
        v2f b = *reinterpret_cast<const v2f*>(
            &Ws[p][(2 * kk + half) * WP_STRIDE + (nc * 16 + ln) * 2]);
        acc[nc] = wmma_f32_16x16x4(a, b, acc[nc]);
      }
    }
  }

  // ---- epilogue: C row = g + 8*half, col = ln (ISA C/D layout) ----
  #pragma unroll
  for (int nc = 0; nc < 4; ++nc) {
    #pragma unroll
    for (int g = 0; g < 8; ++g) {
      int row = rowBase + wr + g + 8 * half;
      int col = colBase + nc * 16 + ln;
      C[(size_t)row * N + col] = acc[nc][g] + bias[col];
    }
  }
}

// ---------------------------------------------------------------------------
// Attention: one wave per (b, h, 16-query block). Flash-style online softmax
// over key blocks 0..qb (causal, triangular work). Computes S^T = K_blk*Q^T so
// per-query softmax stats live per-lane. Self-attention column folded at end.
// qkv layout: [B*S, 3*D]; out layout: [B*S, D].
// ---------------------------------------------------------------------------
__global__ __launch_bounds__(256)
void attn_f32(const float* __restrict__ qkv,
              const float* __restrict__ tk,   // [B,H,S,HD]
              const float* __restrict__ tv,   // [B,H,S,HD]
              float* __restrict__ out) {
  __shared__ float plds[8 * 16 * 17];         // per-wave P bounce, padded

  const int tid  = threadIdx.x;
  const int lane = tid & 31;
  const int wave = tid >> 5;
  const int ln   = lane & 15;
  const int half = lane >> 4;

  const int wid = blockIdx.x * 8 + wave;
  const int qb  = wid & 63;
  const int h   = (wid >> 6) & 15;
  const int b   = wid >> 10;

  const int qrow = qb * 16 + ln;
  const float* qptr  = qkv + (size_t)(b * S_ + qrow) * (3 * D_) + h * HD_;
  const float* ksptr = qptr + D_;
  const float* vbase = qkv + (size_t)b * S_ * (3 * D_) + 2 * D_ + h * HD_;
  const float* tkbase = tk + (size_t)(b * H_ + h) * S_ * HD_;
  const float* tvbase = tv + (size_t)(b * H_ + h) * S_ * HD_;

  v2f qf[16];
  float sdot = 0.f;
  #pragma unroll
  for (int kk = 0; kk < 16; ++kk) {
    const int off = 4 * kk + 2 * half;
    qf[kk] = *reinterpret_cast<const v2f*>(qptr + off);
    v2f ks = *reinterpret_cast<const v2f*>(ksptr + off);
    sdot += qf[kk].x * ks.x + qf[kk].y * ks.y;
  }
  sdot += __shfl_xor(sdot, 16, 32);
  const float s_self = sdot * SCALE_;

  float m = -3.0e38f;
  float l = 0.f;
  v8f acc[4] = {};

  float* myp = &plds[wave * 16 * 17];

  // prefetch first key block (row halves per lane)
  __builtin_prefetch(tkbase + (size_t)ln * HD_ + half * 32, 0, 0);
  __builtin_prefetch(tvbase + (size_t)ln * HD_ + half * 32, 0, 0);

  for (int kb = 0; kb <= qb; ++kb) {
    // prefetch next key block while this one is being consumed
    if (kb < qb) {
      __builtin_prefetch(tkbase + (size_t)((kb + 1) * 16 + ln) * HD_ + half * 32, 0, 0);
      __builtin_prefetch(tvbase + (size_t)((kb + 1) * 16 + ln) * HD_ + half * 32, 0, 0);
    }

    const float* krow = tkbase + (size_t)(kb * 16 + ln) * HD_;
    v2f kf[16];
    #pragma unroll
    for (int kk = 0; kk < 16; ++kk)
      kf[kk] = *reinterpret_cast<const v2f*>(krow + 4 * kk + 2 * half);

    v8f st = {};
    #pragma unroll
    for (int kk = 0; kk < 16; ++kk)
      st = wmma_f32_16x16x4(kf[kk], qf[kk], st);

    #pragma unroll
    for (int g = 0; g < 8; ++g) {
      float s = st[g] * SCALE_;
      if (kb == qb && (g + 8 * half) >= ln) s = MASKED_BIAS_;
      st[g] = s;
    }

    float bm = st[0];
    #pragma unroll
    for (int g = 1; g < 8; ++g) bm = fmaxf(bm, st[g]);
    bm = fmaxf(bm, __shfl_xor(bm, 16, 32));
    const float mnew = fmaxf(m, bm);
    const float corr = __expf(m - mnew);

    float bs = 0.f;
    #pragma unroll
    for (int g = 0; g < 8; ++g) {
      float pexp = __expf(st[g] - mnew);
      st[g] = pexp;
      bs += pexp;
    }
    bs += __shfl_xor(bs, 16, 32);
    l = l * corr + bs;
    m = mnew;

    #pragma unroll
    for (int g = 0; g < 8; ++g) {
      float cr = __shfl(corr, g + 8 * half, 32);
      acc[0][g] *= cr; acc[1][g] *= cr; acc[2][g] *= cr; acc[3][g] *= cr;
    }

    #pragma unroll
    for (int g = 0; g < 8; ++g)
      myp[ln * 17 + g + 8 * half] = st[g];

    const float* tvblk = tvbase + (size_t)kb * 16 * HD_;
    #pragma unroll
    for (int kk2 = 0; kk2 < 4; ++kk2) {
      const int ka = 4 * kk2 + 2 * half;
      v2f a;
      a.x = myp[ln * 17 + ka];
      a.y = myp[ln * 17 + ka + 1];
      #pragma unroll
      for (int nc = 0; nc < 4; ++nc) {
        v2f bf;
        bf.x = tvblk[(size_t)ka * HD_ + nc * 16 + ln];
        bf.y = tvblk[(size_t)(ka + 1) * HD_ + nc * 16 + ln];
        acc[nc] = wmma_f32_16x16x4(a, bf, acc[nc]);
      }
    }
  }

  const float mfin  = fmaxf(m, s_self);
  const float corrf = __expf(m - mfin);
  const float pself = __expf(s_self - mfin);
  const float invl  = 1.0f / (l * corrf + pself);
  const float rowscale = corrf * invl;
  const float wself    = pself * invl;

  #pragma unroll
  for (int g = 0; g < 8; ++g) {
    const int row = qb * 16 + g + 8 * half;
    float rs = __shfl(rowscale, g + 8 * half, 32);
    float ws = __shfl(wself,    g + 8 * half, 32);
    const float* vrow = vbase + (size_t)row * (3 * D_);
    float* orow = out + (size_t)(b * S_ + row) * D_ + h * HD_;
    #pragma unroll
    for (int nc = 0; nc < 4; ++nc) {
      int col = nc * 16 + ln;
      orow[col] = acc[nc][g] * rs + ws * vrow[col];
    }
  }
}

// ---------------------------------------------------------------------------
extern "C" void kernel_launch(void* const* d_in, const int* in_sizes, int n_in,
                              void* d_out, int out_size, void* d_ws, size_t ws_size,
                              hipStream_t stream) {
  const float* hs = (const float*)d_in[0];
  const float* tk = (const float*)d_in[1];
  const float* tv = (const float*)d_in[2];
  const float* Wc = (const float*)d_in[3];
  const float* bc = (const float*)d_in[4];
  const float* Wp = (const float*)d_in[5];
  const float* bp = (const float*)d_in[6];
  float* outp = (float*)d_out;

  float* qkv  = (float*)d_ws;                          // [B*S, 3D] = 48 MB
  float* attn = qkv + (size_t)B_ * S_ * 3 * D_;        // [B*S, D]  = 16 MB

  dim3 blk(256);

  gemm_bias_f32<<<dim3((3 * D_) / BN, (B_ * S_) / BM), blk, 0, stream>>>(
      hs, Wc, bc, qkv, B_ * S_, 3 * D_, D_);

  attn_f32<<<dim3((B_ * H_ * (S_ / 16)) / 8), blk, 0, stream>>>(
      qkv, tk, tv, attn);

  gemm_bias_f32<<<dim3(D_ / BN, (B_ * S_) / BM), blk, 0, stream>>>(
      attn, Wp, bp, outp, B_ * S_, D_, D_);
}